// Generator_22179211117343
// MI455X (gfx1250) — compile-verified
//
#include <hip/hip_runtime.h>
#include <hip/hip_bf16.h>

typedef __attribute__((ext_vector_type(2))) float v2f;
typedef __attribute__((ext_vector_type(8))) float v8f;

#define UNITS  1000
#define EMB    100
#define VOC    32001
#define VOCP   32004           // embT row stride (even, 16B-multiple)
#define BATCH  64
#define SEQ    25
#define ROWS   (BATCH*SEQ)     // 1600, = 100 row-tiles of 16
#define NPAD   112             // EMB padded to 7 N-tiles of 16
#define MROWP  1024            // MmatT row stride (padded from 1000)
#define DKP    104             // dkT row stride (padded from 100)
#define UT     63              // ceil(UNITS/16) unit tiles
#define KSLICES 25
#define KPER    1280           // 25*1280 = 32000, +1 scalar tail column

// ---- f32 WMMA wrapper: D = A(16x4) * B(4x16) + C ----
// A frag: lane<16 holds a[M=lane][k0..k0+1], lane>=16 holds a[M=lane-16][k0+2..k0+3]
// B frag: lane<16 holds b[k0..k0+1][N=lane], lane>=16 holds b[k0+2..k0+3][N=lane-16]
// C frag: VGPR v, lane L: M = v + 8*(L>=16), N = L%16
__device__ __forceinline__ v8f wmma4(float a0, float a1, float b0, float b1, v8f c) {
  v2f a; a.x = a0; a.y = a1;
  v2f b; b.x = b0; b.y = b1;
  return __builtin_amdgcn_wmma_f32_16x16x4_f32(false, a, false, b, (short)0, c,
                                               false, false);
}

__device__ __forceinline__ float sgm(float x) { return 1.f / (1.f + __expf(-x)); }
__device__ __forceinline__ float th(float x)  { float e = __expf(2.f * x); return (e - 1.f) / (e + 1.f); }
__device__ __forceinline__ int   imin(int a, int b) { return a < b ? a : b; }

// ---------- zero-init (MmatT must start at 0 for fp32 atomic accumulation) ----------
__global__ void k_init(float* __restrict__ p, int n) {
  int i = blockIdx.x * blockDim.x + threadIdx.x;
  if (i < n) p[i] = 0.f;
}

// ---------- embT[n][k] = emb[k][n], zero pad rows n>=100 ----------
__global__ void k_trans_emb(const float* __restrict__ emb, float* __restrict__ embT) {
  int idx = blockIdx.x * blockDim.x + threadIdx.x;   // over VOC*NPAD, n fastest
  if (idx >= VOC * NPAD) return;
  int k = idx / NPAD, n = idx % NPAD;                // coalesced reads along n
  embT[(size_t)n * VOCP + k] = (n < EMB) ? emb[(size_t)k * EMB + n] : 0.f;
}

// ---------- dkT[c][k] = dec_kernel[k][c], zero pad cols k>=100 ----------
__global__ void k_trans_dk(const float* __restrict__ dk, float* __restrict__ dkT) {
  int idx = blockIdx.x * blockDim.x + threadIdx.x;   // over 3000*DKP, k fastest
  if (idx >= 3000 * DKP) return;
  int c = idx / DKP, k = idx % DKP;                  // coalesced writes along k
  dkT[idx] = (k < EMB) ? dk[(size_t)k * 3000 + c] : 0.f;
}

// ---------- gather decoder inputs: tokens[row], xmat[row][EMB] ----------
__global__ void k_gather(const int* __restrict__ target, const int* __restrict__ start_tok,
                         const float* __restrict__ emb, int* __restrict__ tokens,
                         float* __restrict__ xmat) {
  int row = blockIdx.x;                 // 0..1599
  int b = row / SEQ, t = row % SEQ;
  int tok = (t == 0) ? start_tok[0] : target[b * SEQ + t - 1];
  if (threadIdx.x == 0) tokens[row] = tok;
  if (threadIdx.x < EMB)
    xmat[row * EMB + threadIdx.x] = emb[(size_t)tok * EMB + threadIdx.x];
}

// ---------- Mb[n] = sum_v fc_b[v] * emb[v][n] ----------
__global__ void k_bias(const float* __restrict__ fcb, const float* __restrict__ emb,
                       float* __restrict__ Mb) {
  __shared__ float red[256];
  int n = blockIdx.x;                   // 0..111
  float s = 0.f;
  if (n < EMB)
    for (int v = threadIdx.x; v < VOC; v += 256) s += fcb[v] * emb[(size_t)v * EMB + n];
  red[threadIdx.x] = s;
  __syncthreads();
  for (int off = 128; off; off >>= 1) {
    if (threadIdx.x < off) red[threadIdx.x] += red[threadIdx.x + off];
    __syncthreads();
  }
  if (threadIdx.x == 0) Mb[n] = (n < EMB) ? red[0] : 0.f;
}

// ---------- MmatT[n][m] = (fc_w @ emb)[m][n]  (K-sliced, atomic combine) ----------
__global__ __launch_bounds__(256) void k_proj(const float* __restrict__ fcw,
                                              const float* __restrict__ embT,
                                              float* __restrict__ MmatT) {
  int w = (blockIdx.x * blockDim.x + threadIdx.x) >> 5;
  if (w >= UT * KSLICES) return;        // wave-uniform
  int mt = w / KSLICES, s = w % KSLICES;
  int lane = threadIdx.x & 31, hi = lane >> 4, n16 = lane & 15;
  int am = imin(mt * 16 + n16, UNITS - 1);      // clamped A row (invalid rows unstored)
  const float* arow = fcw + (size_t)am * VOC;
  const float* brow = embT + (size_t)n16 * VOCP;  // B col = n16 + 16*nt
  v8f acc[7] = {};
  int k0 = s * KPER;
  for (int kk = 0; kk < KPER / 4; ++kk) {
    int k = k0 + kk * 4 + 2 * hi;
    // load phase: 1 NT A-pair + 7 B-pairs, all independent registers
    float a0 = __builtin_nontemporal_load(arow + k);
    float a1 = __builtin_nontemporal_load(arow + k + 1);
    float2 b[7];
#pragma unroll
    for (int nt = 0; nt < 7; ++nt)
      b[nt] = *(const float2*)(brow + (size_t)nt * 16 * VOCP + k);  // 8B aligned
    // compute phase
#pragma unroll
    for (int nt = 0; nt < 7; ++nt)
      acc[nt] = wmma4(a0, a1, b[nt].x, b[nt].y, acc[nt]);
  }
  if (s == KSLICES - 1) {               // scalar tail: K column 32000
#pragma unroll
    for (int nt = 0; nt < 7; ++nt) {
      float fb = brow[(size_t)nt * 16 * VOCP + (VOC - 1)];   // zero in pad cols
#pragma unroll
      for (int v = 0; v < 8; ++v) {
        int m = imin(mt * 16 + v + 8 * hi, UNITS - 1);
        float fa = fcw[(size_t)m * VOC + (VOC - 1)];
        acc[nt][v] += fa * fb;
      }
    }
  }
#pragma unroll
  for (int nt = 0; nt < 7; ++nt) {
    int n = nt * 16 + n16;              // < 112
#pragma unroll
    for (int v = 0; v < 8; ++v) {
      int m = mt * 16 + v + 8 * hi;
      if (m < UNITS) atomicAdd(&MmatT[(size_t)n * MROWP + m], acc[nt][v]);
    }
  }
}

// ---------- fused GRU step from h=0: h1[1600,1000] ----------
__global__ __launch_bounds__(256) void k_gru(const float* __restrict__ xmat,
                                             const int* __restrict__ tokens,
                                             const float* __restrict__ dkT,  // [3000][DKP]
                                             const float* __restrict__ db,   // [2,3000]
                                             float* __restrict__ h1) {
  int w = (blockIdx.x * blockDim.x + threadIdx.x) >> 5;
  if (w >= (ROWS / 16) * UT) return;    // wave-uniform
  int rt = w / UT, ut = w % UT;
  int lane = threadIdx.x & 31, hi = lane >> 4, n16 = lane & 15;
  const float* arow = xmat + (size_t)(rt * 16 + n16) * EMB;
  int u = ut * 16 + n16;                // unit index (B column & bias index)
  int cz = imin(u, UNITS - 1);          // clamped: invalid lanes feed unstored C cols
  const float* bz_row = dkT + (size_t)cz * DKP;
  const float* br_row = dkT + (size_t)(UNITS + cz) * DKP;
  const float* bh_row = dkT + (size_t)(2 * UNITS + cz) * DKP;
  v8f az = {}, ar = {}, ah = {};
  for (int kk = 0; kk < EMB / 4; ++kk) {
    int k = kk * 4 + 2 * hi;
    float2 a  = *(const float2*)(arow + k);     // all 8B aligned
    float2 bz = *(const float2*)(bz_row + k);
    float2 br = *(const float2*)(br_row + k);
    float2 bh = *(const float2*)(bh_row + k);
    az = wmma4(a.x, a.y, bz.x, bz.y, az);
    ar = wmma4(a.x, a.y, br.x, br.y, ar);
    ah = wmma4(a.x, a.y, bh.x, bh.y, ah);
  }
  float b0z = db[cz];
  float b1z = db[3000 + cz];
  float b0r = db[UNITS + cz];
  float b1r = db[3000 + UNITS + cz];
  float b0h = db[2 * UNITS + cz];
  float b1h = db[3000 + 2 * UNITS + cz];
#pragma unroll
  for (int v = 0; v < 8; ++v) {
    int m = rt * 16 + v + 8 * hi;       // global row, < 1600
    int tok = tokens[m];
    float z  = sgm(az[v] + b0z + b1z);
    float r  = sgm(ar[v] + b0r + b1r);
    float hh = th(ah[v] + b0h + r * b1h);
    float h  = (tok != 0) ? (1.f - z) * hh : 0.f;
    if (u < UNITS) h1[(size_t)m * UNITS + u] = h;
  }
}

// ---------- out = h1[1600,1000] @ M + Mb   (M read transposed) ----------
__global__ __launch_bounds__(256) void k_out(const float* __restrict__ h1,
                                             const float* __restrict__ MmatT,
                                             const float* __restrict__ Mb,
                                             float* __restrict__ out) {
  int w = (blockIdx.x * blockDim.x + threadIdx.x) >> 5;
  if (w >= (ROWS / 16) * 7) return;     // wave-uniform
  int mt = w / 7, nt = w % 7;
  int lane = threadIdx.x & 31, hi = lane >> 4, n16 = lane & 15;
  const float* arow = h1 + (size_t)(mt * 16 + n16) * UNITS;
  int n = nt * 16 + n16;                // < 112
  const float* brow = MmatT + (size_t)n * MROWP;
  v8f acc = {};
  for (int kk = 0; kk < UNITS / 4; ++kk) {
    int k = kk * 4 + 2 * hi;
    float2 a = *(const float2*)(arow + k);   // 8B aligned
    float2 b = *(const float2*)(brow + k);   // 8B aligned
    acc = wmma4(a.x, a.y, b.x, b.y, acc);
  }
  float bias = Mb[n];                   // Mb padded to 112 (pad entries are 0)
#pragma unroll
  for (int v = 0; v < 8; ++v) {
    int m = mt * 16 + v + 8 * hi;       // < 1600
    if (n < EMB) out[(size_t)m * EMB + n] = acc[v] + bias;
  }
}

extern "C" void kernel_launch(void* const* d_in, const int* in_sizes, int n_in,
                              void* d_out, int out_size, void* d_ws, size_t ws_size,
                              hipStream_t stream) {
  (void)in_sizes; (void)n_in; (void)out_size; (void)ws_size;
  const int*   target = (const int*)d_in[1];
  const int*   start  = (const int*)d_in[2];
  const float* emb    = (const float*)d_in[3];
  const float* dk     = (const float*)d_in[7];   // dec_kernel
  const float* db     = (const float*)d_in[9];   // dec_bias
  const float* fcw    = (const float*)d_in[10];
  const float* fcb    = (const float*)d_in[11];
  float* out = (float*)d_out;

  char* ws = (char*)d_ws;
  int*   tokens = (int*)  (ws + 0);           //     6,400 B
  float* xmat   = (float*)(ws + 6400);        //   640,000 B
  float* h1     = (float*)(ws + 646400);      // 6,400,000 B
  float* MmatT  = (float*)(ws + 7046400);     //   458,752 B  [112][1024]
  float* Mb     = (float*)(ws + 7505152);     //       448 B
  float* embT   = (float*)(ws + 7505600);     // 14,337,792 B [112][32004]
  float* dkT    = (float*)(ws + 21843392);    // 1,248,000 B  [3000][104]

  k_init<<<(NPAD * MROWP + 255) / 256, 256, 0, stream>>>(MmatT, NPAD * MROWP);
  k_trans_emb<<<(VOC * NPAD + 255) / 256, 256, 0, stream>>>(emb, embT);
  k_trans_dk<<<(3000 * DKP + 255) / 256, 256, 0, stream>>>(dk, dkT);
  k_gather<<<ROWS, 128, 0, stream>>>(target, start, emb, tokens, xmat);
  k_bias<<<NPAD, 256, 0, stream>>>(fcb, emb, Mb);

  int nw1 = UT * KSLICES;                     // 1575 waves
  k_proj<<<(nw1 * 32 + 255) / 256, 256, 0, stream>>>(fcw, embT, MmatT);

  int nw2 = (ROWS / 16) * UT;                 // 6300 waves
  k_gru<<<(nw2 * 32 + 255) / 256, 256, 0, stream>>>(xmat, tokens, dkT, db, h1);

  int nw3 = (ROWS / 16) * 7;                  // 700 waves
  k_out<<<(nw3 * 32 + 255) / 256, 256, 0, stream>>>(h1, MmatT, Mb, out);
}